// SSIM_67997922230741
// MI455X (gfx1250) — compile-verified
//
#include <hip/hip_runtime.h>

// CDNA5 / gfx1250 SSIM kernel.
// Separable 11x11 Gaussian conv expressed as banded-matrix matmuls executed
// with V_WMMA_F32_16X16X4_F32:  OUT(16x16) = Aband(16x26) * S(26x26) * G(26x16)
// for the 5 signals (x, y, x*x, y*y, x*y) simultaneously (5 interleaved
// accumulator chains), then elementwise SSIM and a deterministic reduction.

typedef __attribute__((ext_vector_type(8))) float v8f;
typedef __attribute__((ext_vector_type(2))) float v2f;

#define IMG_H   512
#define IMG_W   512
#define SROWS   28      // 26-row input region zero-padded to 28 (K padded to 28)
#define SCOLS   28      // 26 cols zero-padded to 28
#define HROWS   28      // intermediate H matrix rows (K of vertical pass)
#define HCOLS   17      // 16 cols + 1 pad -> conflict-free B-operand reads
#define NSIG    5       // x, y, x*x, y*y, x*y
#define WPB     4       // waves per block
#define TPW     12      // tiles per wave
#define NBLK    1024    // 1024 blocks * 4 waves * 12 tiles = 49152 = 48*32*32
#define SSIM_C1 (1.0e-4f)
#define SSIM_C2 (9.0e-4f)

__device__ __forceinline__ float bandv(const float* gws, int d) {
  // banded Gaussian matrix entry: g[d] if 0 <= d < 11 else 0
  return ((unsigned)d < 11u) ? gws[d] : 0.0f;
}

__device__ __forceinline__ v8f wmma_f32(v2f a, v2f b, v8f c) {
  return __builtin_amdgcn_wmma_f32_16x16x4_f32(false, a, false, b,
                                               (short)0, c, false, false);
}

__global__ __launch_bounds__(WPB * 32) void ssim_tiles(
    const float* __restrict__ img1, const float* __restrict__ img2,
    const float* __restrict__ win, float* __restrict__ partial) {
  __shared__ float sI1[WPB][SROWS][SCOLS];          // per-wave img1 region
  __shared__ float sI2[WPB][SROWS][SCOLS];          // per-wave img2 region
  __shared__ float sHb[WPB][NSIG][HROWS][HCOLS];    // horizontal-conv results
  __shared__ float sGW[16];                         // 11-tap 1D gaussian
  __shared__ float sRed[WPB];

  const int tid  = threadIdx.x;
  const int lane = tid & 31;
  const int wid  = tid >> 5;
  const int m    = lane & 15;   // M (and N) index for this lane
  const int hi   = lane >> 4;   // lane-half selects K pair within a K=4 chunk

  // Recover the 1D gaussian from the reference's 2D window:
  // w2d[5][j] = g[5]*g[j], w2d[5][5] = g[5]^2 -> g[j] = w2d[5][j]/sqrt(w2d[5][5])
  if (tid < 11) sGW[tid] = win[5 * 11 + tid] / sqrtf(win[5 * 11 + 5]);
  __syncthreads();

  // Banded Gaussian WMMA operand, invariant across signals and tiles:
  // horizontal B operand G[k][n] = g[k-n] and vertical A operand
  // Aband[m][k] = g[k-m] are the same per-lane values (n == m). Keep in VGPRs.
  v2f bw[7];
#pragma unroll
  for (int kc = 0; kc < 7; ++kc) {
    const int k0 = 4 * kc + 2 * hi;
    bw[kc].x = bandv(sGW, k0 - m);
    bw[kc].y = bandv(sGW, k0 + 1 - m);
  }

  float (*I1)[SCOLS] = sI1[wid];
  float (*I2)[SCOLS] = sI2[wid];
  float (*Hb)[HROWS][HCOLS] = sHb[wid];

  const int gwid = blockIdx.x * WPB + wid;
  float wsum = 0.0f;

  for (int t = 0; t < TPW; ++t) {
    const int tile  = gwid * TPW + t;
    const int plane = tile >> 10;          // 48 planes (16 imgs * 3 channels)
    const int ty    = (tile >> 5) & 31;
    const int tx    = tile & 31;
    const float* p1 = img1 + (size_t)plane * (IMG_H * IMG_W);
    const float* p2 = img2 + (size_t)plane * (IMG_H * IMG_W);
    const int y0 = ty * 16 - 5;            // input region origin (zero pad = 5)
    const int x0 = tx * 16 - 5;

    // Fill zero-padded 28x28 regions.
    for (int idx = lane; idx < SROWS * SCOLS; idx += 32) {
      const int r = idx / SCOLS;
      const int c = idx - r * SCOLS;
      float a = 0.0f, b = 0.0f;
      const int gy = y0 + r, gx = x0 + c;
      if (r < 26 && c < 26 && (unsigned)gy < (unsigned)IMG_H &&
          (unsigned)gx < (unsigned)IMG_W) {
        const int off = gy * IMG_W + gx;
        a = p1[off];
        b = p2[off];
      }
      I1[r][c] = a;
      I2[r][c] = b;
    }
    __syncthreads();

    // ---- horizontal pass: H_s(26x16) = S_s(26x26[pad28]) * G(26x16) ----
    // K-chunk outermost: read img values once, feed 5 interleaved WMMA chains.
#pragma unroll
    for (int bb = 0; bb < 2; ++bb) {
      v8f acc[NSIG] = {};
      // A-row for this lane; rows 26/27 are zero pad, so clamping block 1's
      // rows 28..31 onto row 27 keeps them zero.
      const int rowA = min(16 * bb + m, SROWS - 1);
#pragma unroll
      for (int kc = 0; kc < 7; ++kc) {
        const int k0 = 4 * kc + 2 * hi;     // this lane's K pair
        const float i1a = I1[rowA][k0];
        const float i1b = I1[rowA][k0 + 1];
        const float i2a = I2[rowA][k0];
        const float i2b = I2[rowA][k0 + 1];
        v2f A;
        A.x = i1a;        A.y = i1b;        acc[0] = wmma_f32(A, bw[kc], acc[0]);
        A.x = i2a;        A.y = i2b;        acc[1] = wmma_f32(A, bw[kc], acc[1]);
        A.x = i1a * i1a;  A.y = i1b * i1b;  acc[2] = wmma_f32(A, bw[kc], acc[2]);
        A.x = i2a * i2a;  A.y = i2b * i2b;  acc[3] = wmma_f32(A, bw[kc], acc[3]);
        A.x = i1a * i2a;  A.y = i1b * i2b;  acc[4] = wmma_f32(A, bw[kc], acc[4]);
      }
      // D layout: VGPR v -> row v (lanes 0-15) / row v+8 (lanes 16-31)
#pragma unroll
      for (int s = 0; s < NSIG; ++s) {
#pragma unroll
        for (int v = 0; v < 8; ++v) {
          const int row = 16 * bb + v + 8 * hi;
          if (row < HROWS) Hb[s][row][m] = acc[s][v];
        }
      }
    }
    __syncthreads();

    // ---- vertical pass: OUT_s(16x16) = Aband(16x26[pad28]) * H_s(28x16) ----
    v8f outv[NSIG] = {};
#pragma unroll
    for (int kc = 0; kc < 7; ++kc) {
      const int k0 = 4 * kc + 2 * hi;
#pragma unroll
      for (int s = 0; s < NSIG; ++s) {
        v2f B;
        B.x = Hb[s][k0][m];                 // B[k][n], n == m
        B.y = Hb[s][k0 + 1][m];
        outv[s] = wmma_f32(bw[kc], B, outv[s]);
      }
    }
    __syncthreads();  // protect Hb and sI (WAR) before next tile

    // ---- elementwise SSIM on the 8 outputs this lane owns ----
#pragma unroll
    for (int j = 0; j < 8; ++j) {
      const float mu1   = outv[0][j];
      const float mu2   = outv[1][j];
      const float mu1sq = mu1 * mu1;
      const float mu2sq = mu2 * mu2;
      const float mu12  = mu1 * mu2;
      const float s1    = outv[2][j] - mu1sq;
      const float s2    = outv[3][j] - mu2sq;
      const float s12   = outv[4][j] - mu12;
      const float num   = (2.0f * mu12 + SSIM_C1) * (2.0f * s12 + SSIM_C2);
      const float den   = (mu1sq + mu2sq + SSIM_C1) * (s1 + s2 + SSIM_C2);
      wsum += num / den;
    }
  }

  // deterministic reduction: lane-accumulated -> wave shuffle -> block
#pragma unroll
  for (int off = 16; off > 0; off >>= 1) wsum += __shfl_xor(wsum, off, 32);
  if (lane == 0) sRed[wid] = wsum;
  __syncthreads();
  if (tid == 0) {
    float s = 0.0f;
#pragma unroll
    for (int w = 0; w < WPB; ++w) s += sRed[w];
    partial[blockIdx.x] = s;
  }
}

__global__ __launch_bounds__(256) void ssim_reduce(
    const float* __restrict__ partial, float* __restrict__ out) {
  __shared__ float buf[256];
  float s = 0.0f;
  for (int i = threadIdx.x; i < NBLK; i += 256) s += partial[i];
  buf[threadIdx.x] = s;
  __syncthreads();
  for (int off = 128; off > 0; off >>= 1) {
    if (threadIdx.x < off) buf[threadIdx.x] += buf[threadIdx.x + off];
    __syncthreads();
  }
  if (threadIdx.x == 0)
    out[0] = 1.0f - buf[0] * (1.0f / 12582912.0f);  // 16*3*512*512
}

extern "C" void kernel_launch(void* const* d_in, const int* in_sizes, int n_in,
                              void* d_out, int out_size, void* d_ws,
                              size_t ws_size, hipStream_t stream) {
  (void)in_sizes; (void)n_in; (void)out_size; (void)ws_size;
  const float* img1 = (const float*)d_in[0];
  const float* img2 = (const float*)d_in[1];
  const float* win  = (const float*)d_in[2];
  float* partial    = (float*)d_ws;   // NBLK floats = 4 KB
  float* out        = (float*)d_out;

  hipLaunchKernelGGL(ssim_tiles, dim3(NBLK), dim3(WPB * 32), 0, stream,
                     img1, img2, win, partial);
  hipLaunchKernelGGL(ssim_reduce, dim3(1), dim3(256), 0, stream, partial, out);
}